// GCN_24773371363585
// MI455X (gfx1250) — compile-verified
//
#include <hip/hip_runtime.h>
#include <math.h>

#define F_OUT 64

typedef __attribute__((ext_vector_type(2))) float v2f;
typedef __attribute__((ext_vector_type(8))) float v8f;

static inline int cdiv(int a, int b) { return (a + b - 1) / b; }

// ---------------------------------------------------------------- degrees ---
__global__ void k_init_norm(float* __restrict__ ns, float* __restrict__ nd, int n) {
    int i = blockIdx.x * blockDim.x + threadIdx.x;
    if (i < n) { ns[i] = 1.0f; nd[i] = 1.0f; }   // self-loop contributes 1 to each degree
}

__global__ void k_degree(const int* __restrict__ src, const int* __restrict__ dst,
                         float* __restrict__ ns, float* __restrict__ nd, int e) {
    int i = blockIdx.x * blockDim.x + threadIdx.x;
    if (i < e) {
        atomicAdd(&ns[src[i]], 1.0f);
        atomicAdd(&nd[dst[i]], 1.0f);
    }
}

__global__ void k_rsqrt(float* __restrict__ ns, float* __restrict__ nd, int n) {
    int i = blockIdx.x * blockDim.x + threadIdx.x;
    if (i < n) { ns[i] = rsqrtf(ns[i]); nd[i] = rsqrtf(nd[i]); }
}

// ------------------------------------------------- WMMA GEMM + row scaling ---
// H[n,64] = H2[n,64] = (X[n,K] @ W[K,64]) * norm[row].
// H  = per-edge message buffer, H2 = aggregation buffer initialized with the
// self-loop contribution (identical value) -- fused dual store saves a full
// 25.6 MB read+write copy pass per layer.
// One wave per 16-row tile, all 64 columns in 4 f32 16x16 accumulators, using
// V_WMMA_F32_16X16X4_F32 (exact f32, matching the f32 reference).
template <int K>
__global__ __launch_bounds__(256) void k_gemm_scale(const float* __restrict__ X,
                                                    const float* __restrict__ W,
                                                    const float* __restrict__ norm,
                                                    float* __restrict__ H,
                                                    float* __restrict__ H2, int nrows) {
    __shared__ float ldsW[F_OUT * (K + 2)];  // transposed, padded stride vs bank conflicts

    for (int t = threadIdx.x; t < K * F_OUT; t += blockDim.x) {
        int k = t / F_OUT, c = t % F_OUT;         // W[k*64 + c]
        ldsW[c * (K + 2) + k] = W[t];
    }
    __syncthreads();

    const int wave  = threadIdx.x >> 5;
    const int lane  = threadIdx.x & 31;
    const int half  = lane >> 4;                  // 0: K pair {k,k+1}, 1: {k+2,k+3}
    const int l16   = lane & 15;
    const int mBase = (blockIdx.x * 8 + wave) * 16;
    if (mBase >= nrows) return;                   // uniform per wave: EXEC stays all-ones

    int rowA = mBase + l16;
    if (rowA >= nrows) rowA = nrows - 1;          // clamp (loads only; harmless dup)
    const float* xrow = X + (size_t)rowA * K;

    v8f c0 = {}, c1 = {}, c2 = {}, c3 = {};
    for (int k0 = 0; k0 < K; k0 += 4) {
        const int kk = k0 + 2 * half;
        // A (16x4): lane l16 holds row mBase+l16, K = kk, kk+1
        v2f a = *(const v2f*)(xrow + kk);
        // B (4x16): lane l16 holds col tile*16+l16, K = kk, kk+1 (transposed LDS)
        v2f b0 = *(const v2f*)(&ldsW[(0 * 16 + l16) * (K + 2) + kk]);
        v2f b1 = *(const v2f*)(&ldsW[(1 * 16 + l16) * (K + 2) + kk]);
        v2f b2 = *(const v2f*)(&ldsW[(2 * 16 + l16) * (K + 2) + kk]);
        v2f b3 = *(const v2f*)(&ldsW[(3 * 16 + l16) * (K + 2) + kk]);
        c0 = __builtin_amdgcn_wmma_f32_16x16x4_f32(false, a, false, b0, (short)0, c0, false, false);
        c1 = __builtin_amdgcn_wmma_f32_16x16x4_f32(false, a, false, b1, (short)0, c1, false, false);
        c2 = __builtin_amdgcn_wmma_f32_16x16x4_f32(false, a, false, b2, (short)0, c2, false, false);
        c3 = __builtin_amdgcn_wmma_f32_16x16x4_f32(false, a, false, b3, (short)0, c3, false, false);
    }

    // C/D layout: VGPR r, lanes 0-15 -> M=r, lanes 16-31 -> M=r+8; N = l16 (+16*tile)
#pragma unroll
    for (int r = 0; r < 8; ++r) {
        int row = mBase + 8 * half + r;
        if (row < nrows) {
            float s = norm[row];
            size_t base = (size_t)row * F_OUT + l16;
            float v0 = c0[r] * s, v1 = c1[r] * s, v2 = c2[r] * s, v3 = c3[r] * s;
            H [base +  0] = v0;  H [base + 16] = v1;
            H [base + 32] = v2;  H [base + 48] = v3;
            H2[base +  0] = v0;  H2[base + 16] = v1;   // self-loop init of agg buffer
            H2[base + 32] = v2;  H2[base + 48] = v3;
        }
    }
}

// -------------------------------------------------------------- edge scatter ---
// 16 threads per edge, float4 per thread: coalesced 256B gather + 4 f32 atomics.
// Agg buffer (25.6 MB) is L2-resident (192 MB L2) so atomics run at L2 rate.
__global__ void k_scatter(const int* __restrict__ src, const int* __restrict__ dst,
                          const float* __restrict__ h, float* __restrict__ agg, int e) {
    int t = blockIdx.x * blockDim.x + threadIdx.x;
    int edge = t >> 4;
    if (edge >= e) return;
    int c = (t & 15) * 4;
    int s = src[edge], d = dst[edge];
    const float4 v = *(const float4*)(h + (size_t)s * F_OUT + c);
    float* p = agg + (size_t)d * F_OUT + c;
    atomicAdd(p + 0, v.x);
    atomicAdd(p + 1, v.y);
    atomicAdd(p + 2, v.z);
    atomicAdd(p + 3, v.w);
}

// -------------------------------------------------- norm_dst + bias (+ReLU) ---
// float4-vectorized: B128 load/store, 1/4 the threads.
__global__ void k_finalize(const float* __restrict__ agg, const float* __restrict__ nd,
                           const float* __restrict__ bias, float* __restrict__ out,
                           int n, int do_relu) {
    int t = blockIdx.x * blockDim.x + threadIdx.x;   // n*16 threads
    if (t >= n * (F_OUT / 4)) return;
    int row = t >> 4, c4 = (t & 15) * 4;
    float s = nd[row];
    float4 a = *(const float4*)(agg + (size_t)row * F_OUT + c4);
    float4 b = *(const float4*)(bias + c4);
    float4 v;
    v.x = fmaf(a.x, s, b.x);
    v.y = fmaf(a.y, s, b.y);
    v.z = fmaf(a.z, s, b.z);
    v.w = fmaf(a.w, s, b.w);
    if (do_relu) {
        v.x = fmaxf(v.x, 0.0f); v.y = fmaxf(v.y, 0.0f);
        v.z = fmaxf(v.z, 0.0f); v.w = fmaxf(v.w, 0.0f);
    }
    *(float4*)(out + (size_t)row * F_OUT + c4) = v;
}

// ---------------------------------------------------------------------------
extern "C" void kernel_launch(void* const* d_in, const int* in_sizes, int n_in,
                              void* d_out, int out_size, void* d_ws, size_t ws_size,
                              hipStream_t stream) {
    const float* x   = (const float*)d_in[0];
    const int*   src = (const int*)d_in[1];
    const int*   dst = (const int*)d_in[2];
    const float* W1  = (const float*)d_in[3];
    const float* b1  = (const float*)d_in[4];
    const float* W2  = (const float*)d_in[5];
    const float* b2  = (const float*)d_in[6];
    float*       out = (float*)d_out;

    const int n = in_sizes[0] / 128;   // 100000 nodes
    const int e = in_sizes[1];         // 1600000 edges

    float* ws   = (float*)d_ws;
    float* buf0 = ws;                              // [n,64] messages
    float* buf1 = ws + (size_t)n * F_OUT;          // [n,64] aggregation / features
    float* ns   = ws + 2 * (size_t)n * F_OUT;      // [n] norm_src
    float* nd   = ns + n;                          // [n] norm_dst

    const int B = 256;
    const int gemmBlocks = cdiv(cdiv(n, 16), 8);   // 8 waves (16 rows each) per block

    // degrees -> rsqrt norms
    k_init_norm<<<cdiv(n, B), B, 0, stream>>>(ns, nd, n);
    k_degree<<<cdiv(e, B), B, 0, stream>>>(src, dst, ns, nd, e);
    k_rsqrt<<<cdiv(n, B), B, 0, stream>>>(ns, nd, n);

    // ---- layer 1: h = relu(D_dst^-1/2 * A * D_src^-1/2 * (x W1) + b1) ----
    k_gemm_scale<128><<<gemmBlocks, B, 0, stream>>>(x, W1, ns, buf0, buf1, n);
    k_scatter<<<cdiv(e * 16, B), B, 0, stream>>>(src, dst, buf0, buf1, e);
    k_finalize<<<cdiv(n * 16, B), B, 0, stream>>>(buf1, nd, b1, buf1, n, 1);

    // ---- layer 2: out = D_dst^-1/2 * A * D_src^-1/2 * (h W2) + b2 ----
    k_gemm_scale<64><<<gemmBlocks, B, 0, stream>>>(buf1, W2, ns, buf0, out, n);
    k_scatter<<<cdiv(e * 16, B), B, 0, stream>>>(src, dst, buf0, out, e);
    k_finalize<<<cdiv(n * 16, B), B, 0, stream>>>(out, nd, b2, out, n, 0);
}